// NaiveNearestNeighborUpsample_V3_26456998543411
// MI455X (gfx1250) — compile-verified
//
#include <hip/hip_runtime.h>

// Nearest-neighbor 2x upsample, [16,128,128,128] f32 -> [16,128,256,256] f32.
// Pure bandwidth problem: 128 MiB read + 512 MiB write (~27 us at 23.3 TB/s).
// Strategy: each thread reads one float2 of an input row (fully coalesced
// global_load_b64), expands to a float4 {x,x,y,y}, and writes it to BOTH
// corresponding output rows with non-temporal global_store_b128 so the
// write-only 512 MiB stream bypasses/doesn't pollute the 192 MB L2.

typedef __attribute__((ext_vector_type(4))) float f32x4;
typedef __attribute__((ext_vector_type(2))) float f32x2;

// Shape constants from the reference problem instance.
#define PLANES 2048   // B*C = 16*128
#define H_IN   128
#define W_IN   128
// scale_factor = 2 (reference fixes it; out_size/in_size == 4)

__global__ __launch_bounds__(256) void nn_upsample2x_kernel(
    const float* __restrict__ in, float* __restrict__ out) {
  // Flat thread id over (plane, h_in, w_pair); all dims are powers of two.
  const int t  = blockIdx.x * 256 + threadIdx.x;
  const int w2 = t & 63;           // which float2 of the input row   [0,64)
  const int h  = (t >> 6) & 127;   // input row                        [0,128)
  const int p  = t >> 13;          // plane = b*C + c                  [0,2048)

  // Coalesced 8-byte load: lanes of a wave cover 256 contiguous bytes.
  const size_t in_off = ((size_t)p * H_IN + (size_t)h) * W_IN + ((size_t)w2 << 1);
  const f32x2 v = *(const f32x2*)(in + in_off);

  // Horizontal 2x replication: 4 output columns from 2 input columns.
  f32x4 d;
  d.x = v.x; d.y = v.x; d.z = v.y; d.w = v.y;

  // Vertical 2x replication: write the same float4 to output rows 2h, 2h+1.
  // Consecutive lanes write consecutive float4s -> fully coalesced 16B stores.
  const size_t out_row = ((size_t)p * (H_IN * 2) + ((size_t)h << 1)) * (W_IN * 2);
  float* o0 = out + out_row + ((size_t)w2 << 2);

  __builtin_nontemporal_store(d, (f32x4*)o0);                 // row 2h
  __builtin_nontemporal_store(d, (f32x4*)(o0 + (W_IN * 2)));  // row 2h+1
}

extern "C" void kernel_launch(void* const* d_in, const int* in_sizes, int n_in,
                              void* d_out, int out_size, void* d_ws, size_t ws_size,
                              hipStream_t stream) {
  const float* in = (const float*)d_in[0];
  float* out = (float*)d_out;
  (void)in_sizes; (void)n_in; (void)out_size; (void)d_ws; (void)ws_size;

  // One thread per (plane, h_in, w_pair): 2048 * 128 * 64 = 16,777,216 threads.
  const int total  = PLANES * H_IN * (W_IN / 2);
  const int block  = 256;                 // 8 wave32s per block
  const int grid   = total / block;       // exact: 65,536 blocks

  nn_upsample2x_kernel<<<grid, block, 0, stream>>>(in, out);
}